// FlowLenia_34187939676285
// MI455X (gfx1250) — compile-verified
//
#include <hip/hip_runtime.h>
#include <math.h>

#define PL 65536  // one 256x256 plane, in floats

typedef float v2f __attribute__((ext_vector_type(2)));
typedef float v8f __attribute__((ext_vector_type(8)));

__device__ __forceinline__ v8f wmma4(v2f a, v2f b, v8f c) {
  // D = A(16x4 f32) * B(4x16 f32) + C(16x16 f32)
  return __builtin_amdgcn_wmma_f32_16x16x4_f32(false, a, false, b, (short)0, c,
                                               false, false);
}

// -------------------------------------------------------------------------
// 256-point complex FFT per wave via four-step (16x16) with f32 WMMA.
// One vector = 256 complex elements at in[base + n*elem_stride],
// base = (v>>8)*PL + (v&255)*vec_mult.  sign=-1 fwd, +1 inverse.
// -------------------------------------------------------------------------
__global__ void __launch_bounds__(128)
fft256_pass(const float* __restrict__ inr, const float* __restrict__ ini,
            float* __restrict__ outr, float* __restrict__ outi,
            int nvec, int vec_mult, int elem_stride,
            float sign, float scale, int realOnly)
{
  __shared__ float Fc[256], Fs[256];           // 16x16 DFT matrix (cos/sin)
  __shared__ float Xr[4][16][16], Xi[4][16][16];

  const float PI2_16 = 0.39269908169872414f;   // 2*pi/16
  for (int t = threadIdx.x; t < 256; t += blockDim.x) {
    int i = t >> 4, j = t & 15;
    float ang = sign * PI2_16 * (float)((i * j) & 15);
    float sv, cv; __sincosf(ang, &sv, &cv);
    Fc[t] = cv; Fs[t] = sv;
  }
  __syncthreads();

  const int lane = threadIdx.x & 31;
  const int wid  = threadIdx.x >> 5;
  const int lm   = lane & 15;
  const int half = lane >> 4;

  int v = blockIdx.x * (blockDim.x >> 5) + wid;
  if (v >= nvec) return;                       // wave-uniform

  size_t base = (size_t)(v >> 8) * PL + (size_t)(v & 255) * (size_t)vec_mult;

  float (*Mr)[16] = Xr[wid];
  float (*Mi)[16] = Xi[wid];

  // load: M[n1][n2] = x[n1 + 16*n2]
  for (int t = lane; t < 256; t += 32) {
    size_t off = base + (size_t)t * (size_t)elem_stride;
    Mr[t & 15][t >> 4] = inr[off];
    Mi[t & 15][t >> 4] = ini ? ini[off] : 0.0f;
  }

  // ---- step 1: G = M * F16   (complex) ----
  v8f gr = {0,0,0,0,0,0,0,0}, gi = {0,0,0,0,0,0,0,0};
  for (int q = 0; q < 4; ++q) {
    int k0 = 4 * q + 2 * half;                 // K-chunk column base for this lane
    v2f aR = { Mr[lm][k0], Mr[lm][k0 + 1] };
    v2f aI = { Mi[lm][k0], Mi[lm][k0 + 1] };
    v2f bC = { Fc[k0 * 16 + lm], Fc[(k0 + 1) * 16 + lm] };
    v2f bS = { Fs[k0 * 16 + lm], Fs[(k0 + 1) * 16 + lm] };
    v2f bSn = { -bS.x, -bS.y };
    gr = wmma4(aR, bC,  gr);
    gr = wmma4(aI, bSn, gr);
    gi = wmma4(aR, bS,  gi);
    gi = wmma4(aI, bC,  gi);
  }

  // ---- twiddle W256^(n1*k1), store H back to LDS ----
  const float PI2_256 = 0.024543692606170259f; // 2*pi/256
  for (int rI = 0; rI < 8; ++rI) {
    int n1 = rI + 8 * half;                    // D-layout row
    float ang = sign * PI2_256 * (float)((n1 * lm) & 255);
    float sw, cw; __sincosf(ang, &sw, &cw);
    float ar = gr[rI], ai = gi[rI];
    Mr[n1][lm] = ar * cw - ai * sw;
    Mi[n1][lm] = ar * sw + ai * cw;
  }

  // ---- step 3: Z = F16 * H   (complex) ----
  v8f zr = {0,0,0,0,0,0,0,0}, zi = {0,0,0,0,0,0,0,0};
  for (int q = 0; q < 4; ++q) {
    int k0 = 4 * q + 2 * half;
    v2f aC  = { Fc[lm * 16 + k0], Fc[lm * 16 + k0 + 1] };
    v2f aS  = { Fs[lm * 16 + k0], Fs[lm * 16 + k0 + 1] };
    v2f aSn = { -aS.x, -aS.y };
    v2f bR  = { Mr[k0][lm], Mr[k0 + 1][lm] };
    v2f bI  = { Mi[k0][lm], Mi[k0 + 1][lm] };
    zr = wmma4(aC,  bR, zr);
    zr = wmma4(aSn, bI, zr);
    zi = wmma4(aC,  bI, zi);
    zi = wmma4(aS,  bR, zi);
  }

  // ---- store: X[k1 + 16*k2], Z[k2][k1] in D-layout ----
  for (int rI = 0; rI < 8; ++rI) {
    int outn = lm + 16 * rI + 128 * half;
    size_t off = base + (size_t)outn * (size_t)elem_stride;
    if (realOnly) {
      outr[off] = zr[rI] * scale;
    } else {
      outr[off] = zr[rI] * scale;
      outi[off] = zi[rI] * scale;
    }
  }
}

// -------------------------------------------------------------------------
// Spatial kernels, pre-fftshifted (center at index 0).
// -------------------------------------------------------------------------
__global__ void build_kernels_k(const float* __restrict__ R, const float* __restrict__ r,
                                const float* __restrict__ a, const float* __restrict__ b,
                                const float* __restrict__ w, float* __restrict__ ksp)
{
  int idx = blockIdx.x * blockDim.x + threadIdx.x;
  if (idx >= 10 * PL) return;
  int k = idx >> 16;
  int pix = idx & 65535;
  int i = pix >> 8, j = pix & 255;
  float sx = (i < 128) ? (float)i : (float)(i - 256);
  float sy = (j < 128) ? (float)j : (float)(j - 256);
  float D  = sqrtf(sx * sx + sy * sy);
  float Dk = D / ((R[0] + 15.0f) * r[k]);
  float ker = 0.0f;
#pragma unroll
  for (int t = 0; t < 3; ++t) {
    float d = Dk - a[k * 3 + t];
    ker += b[k * 3 + t] * __expf(-d * d / w[k * 3 + t]);
  }
  float sig = 0.5f * (tanhf(-(Dk - 1.0f) * 5.0f) + 1.0f);
  ksp[idx] = sig * ker;
}

__global__ void split_A_k(const float* __restrict__ A,
                          float* __restrict__ ach, float* __restrict__ asum)
{
  int idx = blockIdx.x * blockDim.x + threadIdx.x;
  if (idx >= PL) return;
  float s = 0.0f;
#pragma unroll
  for (int c = 0; c < 3; ++c) {
    float v = A[idx * 3 + c];
    ach[c * PL + idx] = v;
    s += v;
  }
  asum[idx] = s;
}

// fK[k] * fA[C0[k]] / DC(fK[k])   (DC term == spatial kernel sum -> normalization)
__global__ void pointwise_k(const float* __restrict__ fkr, const float* __restrict__ fki,
                            const float* __restrict__ fAr, const float* __restrict__ fAi,
                            float* __restrict__ pr, float* __restrict__ pi)
{
  int idx = blockIdx.x * blockDim.x + threadIdx.x;
  if (idx >= 10 * PL) return;
  int k = idx >> 16;
  int p = idx & 65535;
  int c0 = (k < 9) ? (k / 3) : 0;              // C0 = [0,0,0,1,1,1,2,2,2,0]
  float inv = 1.0f / fkr[(size_t)k * PL];
  float kr = fkr[idx], ki = fki[idx];
  float ar = fAr[c0 * PL + p], ai = fAi[c0 * PL + p];
  pr[idx] = (kr * ar - ki * ai) * inv;
  pi[idx] = (kr * ai + ki * ar) * inv;
}

__global__ void growth_k(const float* __restrict__ U, const float* __restrict__ m,
                         const float* __restrict__ s, const float* __restrict__ h,
                         float* __restrict__ U3)
{
  int idx = blockIdx.x * blockDim.x + threadIdx.x;
  if (idx >= PL) return;
  float acc0 = 0.0f, acc1 = 0.0f, acc2 = 0.0f;
#pragma unroll
  for (int k = 0; k < 10; ++k) {
    float u = U[k * PL + idx];
    float t = (u - m[k]) / s[k];
    float g = (__expf(-0.5f * t * t) * 2.0f - 1.0f) * h[k];
    int ch = (k < 9) ? (k / 3) : 0;            // C1 grouping
    if (ch == 0) acc0 += g; else if (ch == 1) acc1 += g; else acc2 += g;
  }
  U3[0 * PL + idx] = acc0;
  U3[1 * PL + idx] = acc1;
  U3[2 * PL + idx] = acc2;
}

// Sobel (scipy convolve2d 'same', zero fill) + alpha blend -> F planes [d*3+c]
__global__ void compute_F_k(const float* __restrict__ U3, const float* __restrict__ asum,
                            const float* __restrict__ A, float* __restrict__ Fp)
{
  int idx = blockIdx.x * blockDim.x + threadIdx.x;
  if (idx >= PL) return;
  int i = idx >> 8, j = idx & 255;
  const float KXm[3][3] = {{-1.f, 0.f, 1.f}, {-2.f, 0.f, 2.f}, {-1.f, 0.f, 1.f}};
  float gy[3] = {0, 0, 0}, gx[3] = {0, 0, 0}, gya = 0.0f, gxa = 0.0f;
#pragma unroll
  for (int du = -1; du <= 1; ++du) {
#pragma unroll
    for (int dv = -1; dv <= 1; ++dv) {
      int ii = i + du, jj = j + dv;
      if (ii < 0 || ii >= 256 || jj < 0 || jj >= 256) continue;
      float wX = KXm[1 - du][1 - dv];          // convolution: flipped kernel
      float wY = KXm[1 - dv][1 - du];          // KY = KX^T
      float va = asum[ii * 256 + jj];
      gya += va * wY; gxa += va * wX;
#pragma unroll
      for (int c = 0; c < 3; ++c) {
        float vu = U3[c * PL + ii * 256 + jj];
        gy[c] += vu * wY;
        gx[c] += vu * wX;
      }
    }
  }
#pragma unroll
  for (int c = 0; c < 3; ++c) {
    float al = A[idx * 3 + c] * (1.0f / 3.0f);
    al = al * al;
    al = fminf(al, 1.0f);
    Fp[(0 * 3 + c) * PL + idx] = gy[c] * (1.0f - al) - gya * al; // d=0: KY (i-dir)
    Fp[(1 * 3 + c) * PL + idx] = gx[c] * (1.0f - al) - gxa * al; // d=1: KX (j-dir)
  }
}

// Reintegration tracking: 16x16 tile + 5 halo in LDS (9 planes, 24KB).
__global__ void __launch_bounds__(256)
rt_apply_k(const float* __restrict__ A, const float* __restrict__ Fp,
           float* __restrict__ out)
{
  __shared__ float sA[3][26][26];
  __shared__ float sF[6][26][26];
  const int tx = threadIdx.x, ty = threadIdx.y;
  const int tid = ty * 16 + tx;
  const int i0 = (int)blockIdx.y * 16 - 5, j0 = (int)blockIdx.x * 16 - 5;

  for (int t = tid; t < 26 * 26; t += 256) {
    int li = t / 26, lj = t % 26;
    int gi = (i0 + li) & 255, gj = (j0 + lj) & 255;
    int pix = gi * 256 + gj;
    sA[0][li][lj] = A[pix * 3 + 0];
    sA[1][li][lj] = A[pix * 3 + 1];
    sA[2][li][lj] = A[pix * 3 + 2];
#pragma unroll
    for (int p = 0; p < 6; ++p) sF[p][li][lj] = Fp[p * PL + pix];
  }
  __syncthreads();

  int i = (int)blockIdx.y * 16 + ty, j = (int)blockIdx.x * 16 + tx;
  float px = (float)i + 0.5f, py = (float)j + 0.5f;
  const float MA = 4.35f;                      // DD - SIGMA
  const float SG = 0.65f, DTc = 0.2f;
  const float inv4s2 = 1.0f / (4.0f * SG * SG);
  float acc0 = 0.0f, acc1 = 0.0f, acc2 = 0.0f;

  for (int dx = -5; dx <= 5; ++dx) {
    int li = ty + 5 - dx;
    float spx = (float)((i - dx) & 255) + 0.5f;
    for (int dy = -5; dy <= 5; ++dy) {
      int lj = tx + 5 - dy;
      float spy = (float)((j - dy) & 255) + 0.5f;
#pragma unroll
      for (int c = 0; c < 3; ++c) {
        float fx = sF[c][li][lj];
        float fy = sF[3 + c][li][lj];
        float mux = spx + fminf(fmaxf(DTc * fx, -MA), MA);
        float muy = spy + fminf(fmaxf(DTc * fy, -MA), MA);
        mux = fminf(fmaxf(mux, SG), 256.0f - SG);
        muy = fminf(fmaxf(muy, SG), 256.0f - SG);
        float szx = 0.5f + SG - fabsf(px - mux);
        float szy = 0.5f + SG - fabsf(py - muy);
        szx = fminf(fmaxf(szx, 0.0f), 1.0f);   // clip_hi = min(1, 2*sigma) = 1
        szy = fminf(fmaxf(szy, 0.0f), 1.0f);
        float area = szx * szy * inv4s2;
        float av = sA[c][li][lj];
        if (c == 0) acc0 += av * area;
        else if (c == 1) acc1 += av * area;
        else acc2 += av * area;
      }
    }
  }
  int pix = i * 256 + j;
  out[pix * 3 + 0] = acc0;
  out[pix * 3 + 1] = acc1;
  out[pix * 3 + 2] = acc2;
}

// -------------------------------------------------------------------------
extern "C" void kernel_launch(void* const* d_in, const int* in_sizes, int n_in,
                              void* d_out, int out_size, void* d_ws, size_t ws_size,
                              hipStream_t stream)
{
  (void)in_sizes; (void)n_in; (void)out_size; (void)ws_size;
  const float* A = (const float*)d_in[0];
  const float* R = (const float*)d_in[1];
  const float* r = (const float*)d_in[2];
  const float* m = (const float*)d_in[3];
  const float* s = (const float*)d_in[4];
  const float* h = (const float*)d_in[5];
  const float* a = (const float*)d_in[6];
  const float* b = (const float*)d_in[7];
  const float* w = (const float*)d_in[8];
  float* ws  = (float*)d_ws;
  float* out = (float*)d_out;

  // Workspace layout (56 planes = ~14.7 MB, lives in L2)
  float* FKR  = ws + (size_t) 0 * PL;  // 10: kernel FFT real
  float* FKI  = ws + (size_t)10 * PL;  // 10: kernel FFT imag
  float* KSP  = ws + (size_t)20 * PL;  // 10: spatial kernels (temp)
  float* TR   = ws + (size_t)30 * PL;  // 10: row-FFT temp real
  float* TI   = ws + (size_t)40 * PL;  // 10: row-FFT temp imag
  // reuse after kernel FFTs complete:
  float* ACH  = ws + (size_t)20 * PL;  // 3: A channels (planar)
  float* ASUM = ws + (size_t)23 * PL;  // 1: sum over channels
  float* T1R  = ws + (size_t)24 * PL;  // 3
  float* T1I  = ws + (size_t)27 * PL;  // 3
  float* FAR_ = ws + (size_t)30 * PL;  // 3: fft2(A) real
  float* FAI  = ws + (size_t)33 * PL;  // 3: fft2(A) imag
  float* PR   = ws + (size_t)36 * PL;  // 10: products real
  float* PIm  = ws + (size_t)46 * PL;  // 10: products imag
  float* QR   = ws + (size_t) 0 * PL;  // 10: ifft-rows real (reuses FKR)
  float* QI   = ws + (size_t)10 * PL;  // 10: ifft-rows imag (reuses FKI)
  float* U    = ws + (size_t)36 * PL;  // 10: real(ifft2) (reuses PR)
  float* U3   = ws + (size_t)46 * PL;  // 3  (reuses PIm)
  float* FPL  = ws + (size_t)49 * PL;  // 6: flow field planes

  // 1) spatial kernels (already fftshifted)
  build_kernels_k<<<(10 * PL + 255) / 256, 256, 0, stream>>>(R, r, a, b, w, KSP);

  // 2) fft2 of kernels: rows then cols (10 planes -> 2560 vectors per pass)
  fft256_pass<<<640, 128, 0, stream>>>(KSP, nullptr, TR, TI, 2560, 256, 1, -1.0f, 1.0f, 0);
  fft256_pass<<<640, 128, 0, stream>>>(TR, TI, FKR, FKI, 2560, 1, 256, -1.0f, 1.0f, 0);

  // 3) fft2 of A channels
  split_A_k<<<PL / 256, 256, 0, stream>>>(A, ACH, ASUM);
  fft256_pass<<<192, 128, 0, stream>>>(ACH, nullptr, T1R, T1I, 768, 256, 1, -1.0f, 1.0f, 0);
  fft256_pass<<<192, 128, 0, stream>>>(T1R, T1I, FAR_, FAI, 768, 1, 256, -1.0f, 1.0f, 0);

  // 4) spectral product with folded kernel normalization
  pointwise_k<<<(10 * PL) / 256, 256, 0, stream>>>(FKR, FKI, FAR_, FAI, PR, PIm);

  // 5) ifft2, keep real part only, scale 1/65536
  fft256_pass<<<640, 128, 0, stream>>>(PR, PIm, QR, QI, 2560, 256, 1, 1.0f, 1.0f, 0);
  fft256_pass<<<640, 128, 0, stream>>>(QR, QI, U, nullptr, 2560, 1, 256, 1.0f,
                                       1.0f / 65536.0f, 1);

  // 6) growth + h weighting + C1 channel sums
  growth_k<<<PL / 256, 256, 0, stream>>>(U, m, s, h, U3);

  // 7) Sobel + alpha blend -> F
  compute_F_k<<<PL / 256, 256, 0, stream>>>(U3, ASUM, A, FPL);

  // 8) reintegration tracking
  dim3 grd(16, 16), blk(16, 16);
  rt_apply_k<<<grd, blk, 0, stream>>>(A, FPL, out);
}